// SAP_41764261986381
// MI455X (gfx1250) — compile-verified
//
#include <hip/hip_runtime.h>
#include <hip/hip_bf16.h>

// irfft-as-GEMM on CDNA5 WMMA (fp32, V_WMMA_F32_16X16X4_F32).
// out[m,t] = sum_j A[m,j] * B[j,t], K = 32768.
//   A[m,j] = packed allpass half-spectrum coefficients (closed form from alpha,d)
//   B[j,t] = Fourier basis, generated on the fly (materialized B would be 4 GB)
// v2: fragment-ready paired LDS layout (one ds_load_b64 per WMMA operand,
//     no repack movs) + 256x128 output tile (better gen:MAC amortization).

typedef float v2f __attribute__((ext_vector_type(2)));
typedef float v8f __attribute__((ext_vector_type(8)));

#define NSAMP   32768
#define NHALF   16384
#define NFLAT   2048               // 32 * 64 series
#define M_TILE  256
#define N_TILE  128
#define K_CHUNK 32
#define TPB     256

// Pair-row strides in floats. Row holds M_TILE (or N_TILE) float2 pairs plus
// 16 pad pairs so consecutive pair-rows differ by 32 banks (mod 64):
// lanes 0-15 read pair-row p, lanes 16-31 read pair-row p+1 conflict-free.
#define A_STR   ((M_TILE + 16) * 2)   // 544 floats
#define B_STR   ((N_TILE + 16) * 2)   // 288 floats

// H(e^{j 2*pi*k/N}) of (a + z^-d)/(1 + a z^-d). Phase d*w reduced mod 2*pi:
// integer part of d reduced exactly with & (N-1); v_sin/v_cos eat revolutions.
__device__ __forceinline__ void sap_H(float alpha, float d, int k,
                                      float& hr, float& hi) {
  float di = floorf(d);
  float df = d - di;
  int   kdi = (k * (int)di) & (NSAMP - 1);       // exact: k*di < 2^24
  float t = (float)kdi + (float)k * df;          // < 49152
  if (t >= (float)NSAMP) t -= (float)NSAMP;      // [0, NSAMP)
  float x = t * (1.0f / (float)NSAMP);           // revolutions
  float s = __builtin_amdgcn_sinf(x);            // v_sin_f32
  float c = __builtin_amdgcn_cosf(x);            // v_cos_f32
  // e^{-i w d} = c - i s
  float nr  = alpha + c;
  float ni  = -s;
  float dr  = 1.0f + alpha * c;
  float dmi = -alpha * s;
  float inv = __builtin_amdgcn_rcpf(dr * dr + dmi * dmi);
  hr = (nr * dr + ni * dmi) * inv;
  hi = (ni * dr - nr * dmi) * inv;
}

// Packed coefficient row: irfft(H)[t] = Hr0/n + (-1)^t Hr_nyq/n
//   + sum_{k=1}^{n/2-1} (2Hr_k/n) cos(w_k t) - (2Hi_k/n) sin(w_k t)
__device__ __forceinline__ float a_coef(float alpha, float d, int j) {
  const float invn = 1.0f / (float)NSAMP;
  float hr, hi;
  if (j <= 1) {                                  // j=0 DC, j=1 Nyquist
    sap_H(alpha, d, (j == 0) ? 0 : NHALF, hr, hi);
    return hr * invn;
  }
  sap_H(alpha, d, j >> 1, hr, hi);
  return (j & 1) ? (-2.0f * invn * hi) : (2.0f * invn * hr);
}

// Fourier basis element; (k*t) & 32767 is an exact integer phase.
__device__ __forceinline__ float b_basis(int j, int t) {
  if (j == 1) return (t & 1) ? -1.0f : 1.0f;     // cos(pi*t)
  unsigned k = (unsigned)(j >> 1);
  unsigned p = (k * (unsigned)t) & (unsigned)(NSAMP - 1);
  float x = (float)p * (1.0f / (float)NSAMP);
  return (j & 1) ? __builtin_amdgcn_sinf(x) : __builtin_amdgcn_cosf(x);
}

__global__ __launch_bounds__(TPB)
void SAP_irfft_wmma(const float* __restrict__ alphas,
                    const float* __restrict__ ds,
                    float* __restrict__ out) {
  // Paired layout: element (k, x) lives at [(k>>1)*STR + 2*x + (k&1)],
  // so {k even, k odd} for one x are adjacent -> one ds_load_b64 fragment.
  __shared__ __align__(16) float sA[(K_CHUNK / 2) * A_STR];
  __shared__ __align__(16) float sB[(K_CHUNK / 2) * B_STR];
  __shared__ float sAl[M_TILE];
  __shared__ float sD[M_TILE];

  const int tid  = threadIdx.x;
  const int lane = tid & 31;            // wave32
  const int wid  = tid >> 5;            // 0..7
  const int wm   = wid >> 2;            // 0..1 : 128-row strip
  const int wn   = wid & 3;             // 0..3 : 32-col strip
  const int half = lane >> 4;           // upper half-wave holds K pair {2,3}
  const int l16  = lane & 15;

  const int m0 = blockIdx.y * M_TILE;
  const int n0 = blockIdx.x * N_TILE;

  for (int i = tid; i < M_TILE; i += TPB) {
    sAl[i] = alphas[m0 + i];
    sD[i]  = ds[m0 + i];
  }

  v8f acc[8][2];
#pragma unroll
  for (int mi = 0; mi < 8; ++mi)
#pragma unroll
    for (int ni = 0; ni < 2; ++ni)
      acc[mi][ni] = v8f{};

  for (int j0 = 0; j0 < NSAMP; j0 += K_CHUNK) {
    __syncthreads();
    // A tile: K_CHUNK x M_TILE = 8192 elems, m fastest (stride-2 float writes
    // cover 32 distinct banks across a wave).
    for (int i = tid; i < K_CHUNK * M_TILE; i += TPB) {
      int m = i & (M_TILE - 1);
      int k = i >> 8;                   // log2(M_TILE)
      sA[(k >> 1) * A_STR + 2 * m + (k & 1)] = a_coef(sAl[m], sD[m], j0 + k);
    }
    // B tile: K_CHUNK x N_TILE = 4096 elems.
    for (int i = tid; i < K_CHUNK * N_TILE; i += TPB) {
      int nn = i & (N_TILE - 1);
      int k  = i >> 7;                  // log2(N_TILE)
      sB[(k >> 1) * B_STR + 2 * nn + (k & 1)] = b_basis(j0 + k, n0 + nn);
    }
    __syncthreads();

#pragma unroll
    for (int kk = 0; kk < K_CHUNK; kk += 4) {
      const int kq = (kk >> 1) + half;  // pair-row: lanes>=16 take next pair
      v2f bf[2];
#pragma unroll
      for (int ni = 0; ni < 2; ++ni) {
        int col = wn * 32 + ni * 16 + l16;
        bf[ni] = *reinterpret_cast<const v2f*>(&sB[kq * B_STR + 2 * col]);
      }
#pragma unroll
      for (int mi = 0; mi < 8; ++mi) {
        int row = wm * 128 + mi * 16 + l16;
        v2f af = *reinterpret_cast<const v2f*>(&sA[kq * A_STR + 2 * row]);
#pragma unroll
        for (int ni = 0; ni < 2; ++ni) {
          // (neg_a, A, neg_b, B, c_mod, C, reuse_a, reuse_b)
          acc[mi][ni] = __builtin_amdgcn_wmma_f32_16x16x4_f32(
              false, af, false, bf[ni], (short)0, acc[mi][ni], false, false);
        }
      }
    }
  }

  // C/D 16x16 f32 layout: VGPR v -> row v (lanes 0-15) / row v+8 (lanes 16-31)
#pragma unroll
  for (int mi = 0; mi < 8; ++mi) {
#pragma unroll
    for (int ni = 0; ni < 2; ++ni) {
      int col   = n0 + wn * 32 + ni * 16 + l16;
      int rbase = m0 + wm * 128 + mi * 16 + half * 8;
#pragma unroll
      for (int v = 0; v < 8; ++v) {
        out[(size_t)(rbase + v) * (size_t)NSAMP + (size_t)col] = acc[mi][ni][v];
      }
    }
  }
}

extern "C" void kernel_launch(void* const* d_in, const int* in_sizes, int n_in,
                              void* d_out, int out_size, void* d_ws, size_t ws_size,
                              hipStream_t stream) {
  const float* alphas = (const float*)d_in[0];  // [32,64] flat = 2048
  const float* ds     = (const float*)d_in[1];  // [32,64] flat = 2048
  float* out = (float*)d_out;                   // [2048, 32768] f32

  static_assert(NSAMP % N_TILE == 0 && NFLAT % M_TILE == 0, "tiling");
  dim3 grid(NSAMP / N_TILE, NFLAT / M_TILE);    // 256 x 8 workgroups
  SAP_irfft_wmma<<<grid, TPB, 0, stream>>>(alphas, ds, out);
  (void)in_sizes; (void)n_in; (void)out_size; (void)d_ws; (void)ws_size;
}